// SelfShader_9921374454199
// MI455X (gfx1250) — compile-verified
//
#include <hip/hip_runtime.h>

// sigmoid_alpha_blend (pytorch3d) forward, MI455X / gfx1250.
//
// Shapes: zbuf/dists/pix_to_face [N=8,H=512,W=512,K=16], out [N,H,W,4] f32.
// Pure HBM-streaming kernel (~416 MiB moved, ~18 us at 23.3 TB/s); no matrix
// op exists in this computation, so no WMMA — the win is coalesced b128
// non-temporal vector memory ops and cheap transcendental math.
//
// Per the harness contract, the integer input (pix_to_face) is cast to
// const int* (int32); only its sign is consumed.

typedef float vf4 __attribute__((ext_vector_type(4)));
typedef int   vi4 __attribute__((ext_vector_type(4)));

#define K_FACES 16
// log2(e) / sigma, sigma = 1e-4
#define INV_SIGMA_LOG2E (1.4426950408889634f * 1.0e4f)

// 1 - prob_k = (f >= 0) ? sigmoid(d / sigma) : 1
// sigmoid(x) = 1 / (1 + exp2(-x * log2e))
__device__ __forceinline__ float one_minus_prob(float d, int f) {
    float e = __builtin_amdgcn_exp2f(-d * INV_SIGMA_LOG2E); // v_exp_f32
    float s = __builtin_amdgcn_rcpf(1.0f + e);              // v_rcp_f32
    return (f >= 0) ? s : 1.0f;
}

__global__ __launch_bounds__(256) void sigmoid_alpha_blend_kernel(
    const float* __restrict__ zbuf,
    const float* __restrict__ dists,
    const int*   __restrict__ p2f,
    float*       __restrict__ out,
    int npix) {
    int p = blockIdx.x * blockDim.x + threadIdx.x;
    if (p >= npix) return;

    size_t base = (size_t)p * K_FACES;
    const vf4* __restrict__ dv = (const vf4*)(dists + base);
    const vi4* __restrict__ fv = (const vi4*)(p2f   + base);

    // Streamed once; >L2 footprint -> non-temporal loads (th:NT).
    vf4 d0 = __builtin_nontemporal_load(dv + 0);
    vf4 d1 = __builtin_nontemporal_load(dv + 1);
    vf4 d2 = __builtin_nontemporal_load(dv + 2);
    vf4 d3 = __builtin_nontemporal_load(dv + 3);
    vi4 f0 = __builtin_nontemporal_load(fv + 0);
    vi4 f1 = __builtin_nontemporal_load(fv + 1);
    vi4 f2 = __builtin_nontemporal_load(fv + 2);
    vi4 f3 = __builtin_nontemporal_load(fv + 3);
    float z0 = __builtin_nontemporal_load(zbuf + base); // k = 0 only

    float w[K_FACES];
    #pragma unroll
    for (int i = 0; i < 4; ++i) {
        w[0 * 4 + i] = one_minus_prob(d0[i], f0[i]);
        w[1 * 4 + i] = one_minus_prob(d1[i], f1[i]);
        w[2 * 4 + i] = one_minus_prob(d2[i], f2[i]);
        w[3 * 4 + i] = one_minus_prob(d3[i], f3[i]);
    }

    // Balanced product tree: dependency depth 4 instead of 16.
    #pragma unroll
    for (int stride = K_FACES / 2; stride >= 1; stride >>= 1) {
        #pragma unroll
        for (int i = 0; i < stride; ++i) {
            w[i] = w[i] * w[i + stride];
        }
    }
    float alpha = 1.0f - w[0];

    vf4 o;
    o[0] = z0;
    o[1] = z0;
    o[2] = z0;
    o[3] = alpha;
    __builtin_nontemporal_store(o, (vf4*)out + p);
}

extern "C" void kernel_launch(void* const* d_in, const int* in_sizes, int n_in,
                              void* d_out, int out_size, void* d_ws, size_t ws_size,
                              hipStream_t stream) {
    const float* zbuf  = (const float*)d_in[0];
    const float* dists = (const float*)d_in[1];
    const int*   p2f   = (const int*)  d_in[2];
    float*       out   = (float*)d_out;

    int npix = in_sizes[0] / K_FACES; // N*H*W = 2,097,152
    int block = 256;                  // 8 wave32s per block
    int grid  = (npix + block - 1) / block;
    sigmoid_alpha_blend_kernel<<<grid, block, 0, stream>>>(zbuf, dists, p2f, out, npix);
}